// NAMAT_83434034692778
// MI455X (gfx1250) — compile-verified
//
#include <hip/hip_runtime.h>
#include <math.h>

// ---------------------------------------------------------------------------
// CDNA5 (gfx1250) implementation of the gated-GNN reference.
// Heavy GEMMs use V_WMMA_F32_16X16X4_F32 (exact fp32 math, wave32).
// Weights are pre-packed into pair-interleaved layout so each B fragment is a
// single b64 load; A rows are prefetched with global_prefetch_b8.
// ---------------------------------------------------------------------------

typedef float v2f __attribute__((ext_vector_type(2)));
typedef float v8f __attribute__((ext_vector_type(8)));

#define N_NODES 20000
#define K_REL   4
#define N_EDGE  320000
#define D_HID   128
#define HID_IN  256
#define IN_DIM  1024
#define TEMP_F  0.6f
#define M_CLAMP 20.0f

static inline int cdiv_i(int a, int b) { return (a + b - 1) / b; }

// ---------------------------------------------------------------------------
// utility kernels
// ---------------------------------------------------------------------------
__global__ void zero_f32(float* __restrict__ p, long n) {
  long i = (long)blockIdx.x * blockDim.x + threadIdx.x;
  if (i < n) p[i] = 0.0f;
}

__global__ void clamp_f32(float* __restrict__ p, long n, float lo, float hi) {
  long i = (long)blockIdx.x * blockDim.x + threadIdx.x;
  if (i < n) p[i] = fminf(fmaxf(p[i], lo), hi);
}

// Pack W[Kd,Nc] (row-major) into pair-interleaved layout:
//   P[(p*Nc + c)*2 + 0] = W[2p  ][c]
//   P[(p*Nc + c)*2 + 1] = W[2p+1][c]
// so a WMMA B fragment {W[k0+2g][c], W[k0+2g+1][c]} is one contiguous b64.
__global__ void pack_w_pairs(const float* __restrict__ W, float* __restrict__ P,
                             int Kd, int Nc) {
  const int idx = blockIdx.x * blockDim.x + threadIdx.x;
  const int tot = (Kd / 2) * Nc;
  if (idx >= tot) return;
  const int r = idx / Nc;
  const int c = idx - r * Nc;
  P[2 * idx + 0] = W[(2 * r + 0) * Nc + c];
  P[2 * idx + 1] = W[(2 * r + 1) * Nc + c];
}

// ---------------------------------------------------------------------------
// WMMA fp32 GEMM:  C[M,Nc] = act(A[M,Kd] @ W[Kd,Nc] + bias),  Nc = NCT*16.
// One wave computes MT 16-row tiles x Nc columns, so B fragments (one b64 each
// from the packed weights) are amortized over MT WMMAs and A fragments over
// NCT WMMAs.
//
// V_WMMA_F32_16X16X4_F32 operand layout (wave32):
//  A (16x4):  lanes 0-15 rows 0..15; VGPR0 = K(0|2), VGPR1 = K(1|3)
//             (lanes 16-31 carry K+2)  -> per-lane contiguous float2 load.
//  B (4x16):  VGPR0 = row k0+2g, VGPR1 = row k0+1+2g, col = lane&15
//             -> one float2 from the pair-packed weights.
//  C (16x16): VGPR i = rows i (lanes 0-15) and i+8 (lanes 16-31), col = lane&15.
// ---------------------------------------------------------------------------
template<int NCT, int MT, bool RELU, bool BIAS>
__global__ __launch_bounds__(256)
void gemm_wmma_f32(const float* __restrict__ A, const float* __restrict__ Wp,
                   const float* __restrict__ bias, float* __restrict__ C,
                   int Mrows, int Kd) {
  const int Nc   = NCT * 16;
  const int wave = threadIdx.x >> 5;
  const int lane = threadIdx.x & 31;
  const int rt   = blockIdx.x * 8 + wave;     // 8 waves / block
  const int row0 = rt * (MT * 16);
  if (row0 >= Mrows) return;                  // wave-uniform exit (EXEC all-1)

  const int grp = lane >> 4;                  // 0: lanes 0-15, 1: lanes 16-31
  const int lr  = lane & 15;
  const v2f* WP = (const v2f*)Wp;

  v8f acc[MT][NCT] = {};
  const float* arow[MT];
#pragma unroll
  for (int m = 0; m < MT; ++m)
    arow[m] = A + (size_t)(row0 + m * 16 + lr) * Kd + 2 * grp;

  for (int k0 = 0; k0 < Kd; k0 += 4) {
    if (((k0 & 63) == 0) && (k0 + 64 < Kd)) {   // stream-ahead on the A rows
#pragma unroll
      for (int m = 0; m < MT; ++m)
        __builtin_prefetch(arow[m] + k0 + 64, 0, 0);   // global_prefetch_b8
    }
    v2f a[MT];
#pragma unroll
    for (int m = 0; m < MT; ++m) a[m] = *(const v2f*)(arow[m] + k0);
    const v2f* wrow = WP + (size_t)(k0 / 2 + grp) * Nc + lr;
#pragma unroll
    for (int j = 0; j < NCT; ++j) {
      const v2f b = wrow[j * 16];
#pragma unroll
      for (int m = 0; m < MT; ++m)
        acc[m][j] = __builtin_amdgcn_wmma_f32_16x16x4_f32(
            false, a[m], false, b, (short)0, acc[m][j], false, false);
    }
  }

#pragma unroll
  for (int m = 0; m < MT; ++m) {
#pragma unroll
    for (int j = 0; j < NCT; ++j) {
      const int col = j * 16 + lr;
      const float bv = BIAS ? bias[col] : 0.0f;
#pragma unroll
      for (int i = 0; i < 8; ++i) {
        float v = acc[m][j][i] + bv;
        if (RELU) v = fmaxf(v, 0.0f);
        C[(size_t)(row0 + m * 16 + i + 8 * grp) * Nc + col] = v;
      }
    }
  }
}

// ---------------------------------------------------------------------------
// Sparse propagation: for each edge (r,c) of relation k: M[r,k,:] += Z[c,:]
// One wave per edge; Z row read as coalesced float4, scatter via f32 atomics
// (Z=10MB and M=41MB both stay resident in the 192MB L2).
// ---------------------------------------------------------------------------
__global__ __launch_bounds__(256)
void edge_prop(const int* __restrict__ rows, const int* __restrict__ cols,
               const float* __restrict__ Z, float* __restrict__ Mb, int E) {
  const int w    = (int)(((long)blockIdx.x * blockDim.x + threadIdx.x) >> 5);
  const int lane = threadIdx.x & 31;
  const int total = K_REL * E;
  if (w >= total) return;
  const int k = w / E;
  const int e = w - k * E;
  const int r = rows[k * E + e];
  const int c = cols[k * E + e];
  const float4 z = ((const float4*)(Z + (size_t)c * D_HID))[lane];
  float* mp = Mb + ((size_t)r * K_REL + k) * D_HID + lane * 4;
  atomicAdd(mp + 0, z.x);
  atomicAdd(mp + 1, z.y);
  atomicAdd(mp + 2, z.z);
  atomicAdd(mp + 3, z.w);
}

// ---------------------------------------------------------------------------
// Gate score: score[n,k] = relu(Gh[n]+Gm[n,k]+logdeg*wlast+bg1) . Wg2 + bg2
// One wave per (n,k); 4x coalesced 128B loads; wave-level shuffle reduction.
// ---------------------------------------------------------------------------
__global__ __launch_bounds__(256)
void gate_score(const float* __restrict__ Gh, const float* __restrict__ Gm,
                const float* __restrict__ logdeg, const float* __restrict__ wlast,
                const float* __restrict__ bg1, const float* __restrict__ Wg2,
                const float* __restrict__ bg2, float* __restrict__ score) {
  const int w    = (int)(((long)blockIdx.x * blockDim.x + threadIdx.x) >> 5);
  const int lane = threadIdx.x & 31;
  if (w >= N_NODES * K_REL) return;
  const int n = w / K_REL;
  const float ld = logdeg[w];
  const float* gh = Gh + (size_t)n * D_HID;
  const float* gm = Gm + (size_t)w * D_HID;
  float acc = 0.0f;
#pragma unroll
  for (int i = 0; i < 4; ++i) {
    const int d = lane + 32 * i;
    float t = gh[d] + gm[d] + ld * wlast[d] + bg1[d];
    t = fmaxf(t, 0.0f);
    acc += t * Wg2[d];
  }
#pragma unroll
  for (int off = 16; off > 0; off >>= 1) acc += __shfl_down(acc, off, 32);
  if (lane == 0) score[w] = acc + bg2[0];
}

// ---------------------------------------------------------------------------
// Softmax over K=4 with mask + double renorm + floor, per node.
// ---------------------------------------------------------------------------
__global__ void softmax_renorm(const float* __restrict__ score,
                               const float* __restrict__ mask,
                               float* __restrict__ alpha) {
  const int n = blockIdx.x * blockDim.x + threadIdx.x;
  if (n >= N_NODES) return;
  float t[K_REL], a[K_REL];
  float m = -1e30f;
#pragma unroll
  for (int k = 0; k < K_REL; ++k) { t[k] = score[n * K_REL + k] * (1.0f / TEMP_F); m = fmaxf(m, t[k]); }
  float s = 0.0f;
#pragma unroll
  for (int k = 0; k < K_REL; ++k) { a[k] = __expf(t[k] - m); s += a[k]; }
  float s2 = 0.0f;
#pragma unroll
  for (int k = 0; k < K_REL; ++k) { a[k] = (a[k] / s) * mask[n * K_REL + k]; s2 += a[k]; }
  s2 = fmaxf(s2, 1e-12f);
  float s3 = 0.0f;
#pragma unroll
  for (int k = 0; k < K_REL; ++k) { a[k] = fmaxf(a[k] / s2, 1e-8f); s3 += a[k]; }
  s3 = fmaxf(s3, 1e-12f);
#pragma unroll
  for (int k = 0; k < K_REL; ++k) alpha[n * K_REL + k] = a[k] / s3;
}

// ---------------------------------------------------------------------------
// fused = sum_k alpha[n,k]*M[n,k,:];  Y = fused + H  (pre-PairNorm residual)
// ---------------------------------------------------------------------------
__global__ void fused_residual(const float* __restrict__ alpha,
                               const float* __restrict__ Mb,
                               const float* __restrict__ H,
                               float* __restrict__ Y) {
  const long idx = (long)blockIdx.x * blockDim.x + threadIdx.x;
  if (idx >= (long)N_NODES * D_HID) return;
  const int n = (int)(idx >> 7);
  const int d = (int)(idx & 127);
  const float* a = alpha + (size_t)n * K_REL;
  const float* m = Mb + (size_t)n * K_REL * D_HID + d;
  float y = H[idx];
  y += a[0] * m[0] + a[1] * m[D_HID] + a[2] * m[2 * D_HID] + a[3] * m[3 * D_HID];
  Y[idx] = y;
}

// ---------------------------------------------------------------------------
// PairNorm stats: colsum[d] (-> stats[0..127]) and total sum-of-squares
// (-> stats[128]). 128 threads/block, CH rows per block.
// ---------------------------------------------------------------------------
__global__ __launch_bounds__(128)
void colstats(const float* __restrict__ Y, float* __restrict__ stats, int CH) {
  __shared__ float red[128];
  const int d  = threadIdx.x;
  const int r0 = blockIdx.x * CH;
  const int r1 = min(r0 + CH, N_NODES);
  float s = 0.0f, s2 = 0.0f;
  for (int r = r0; r < r1; ++r) {
    const float y = Y[(size_t)r * D_HID + d];
    s += y; s2 += y * y;
  }
  atomicAdd(&stats[d], s);
  red[d] = s2;
  __syncthreads();
  for (int off = 64; off > 0; off >>= 1) {
    if (d < off) red[d] += red[d + off];
    __syncthreads();
  }
  if (d == 0) atomicAdd(&stats[128], red[0]);
}

// stats[d] <- mu_d ; stats[129] <- 1/(sqrt(mean ||x-mu||^2) + 1e-6)
__global__ __launch_bounds__(128)
void finalize_stats(float* __restrict__ stats) {
  __shared__ float red[128];
  const int d = threadIdx.x;
  const float mu = stats[d] * (1.0f / (float)N_NODES);
  red[d] = mu * mu;
  __syncthreads();
  for (int off = 64; off > 0; off >>= 1) {
    if (d < off) red[d] += red[d + off];
    __syncthreads();
  }
  if (d == 0) {
    float ms = stats[128] * (1.0f / (float)N_NODES) - red[0];
    ms = fmaxf(ms, 0.0f);
    stats[129] = 1.0f / (sqrtf(ms) + 1e-6f);
  }
  stats[d] = mu;
}

// H = relu((Y - mu) * inv_norm)
__global__ void norm_relu(const float* __restrict__ Y,
                          const float* __restrict__ stats,
                          float* __restrict__ H) {
  const long idx = (long)blockIdx.x * blockDim.x + threadIdx.x;
  if (idx >= (long)N_NODES * D_HID) return;
  const int d = (int)(idx & 127);
  const float x = (Y[idx] - stats[d]) * stats[129];
  H[idx] = fmaxf(x, 0.0f);
}

// logits[n] = HH[n] . Wh2 + bh2  (one wave per node)
__global__ __launch_bounds__(256)
void dot_out(const float* __restrict__ HH, const float* __restrict__ Wh2,
             const float* __restrict__ bh2, float* __restrict__ out) {
  const int w    = (int)(((long)blockIdx.x * blockDim.x + threadIdx.x) >> 5);
  const int lane = threadIdx.x & 31;
  if (w >= N_NODES) return;
  const float* h = HH + (size_t)w * D_HID;
  float acc = 0.0f;
#pragma unroll
  for (int i = 0; i < 4; ++i) {
    const int d = lane + 32 * i;
    acc += h[d] * Wh2[d];
  }
#pragma unroll
  for (int off = 16; off > 0; off >>= 1) acc += __shfl_down(acc, off, 32);
  if (lane == 0) out[w] = acc + bh2[0];
}

// ---------------------------------------------------------------------------
// host launch
// ---------------------------------------------------------------------------
extern "C" void kernel_launch(void* const* d_in, const int* in_sizes, int n_in,
                              void* d_out, int out_size, void* d_ws, size_t ws_size,
                              hipStream_t stream) {
  (void)in_sizes; (void)n_in; (void)out_size; (void)ws_size;

  const float* X      = (const float*)d_in[0];
  const int*   rows   = (const int*)d_in[1];
  const int*   cols   = (const int*)d_in[2];
  const float* mask   = (const float*)d_in[3];
  const float* logdeg = (const float*)d_in[4];
  const float* Wi0 = (const float*)d_in[5];  const float* bi0 = (const float*)d_in[6];
  const float* Wi1 = (const float*)d_in[7];  const float* bi1 = (const float*)d_in[8];
  const float* Wi2 = (const float*)d_in[9];  const float* bi2 = (const float*)d_in[10];
  const float* Wmsg[2] = {(const float*)d_in[11], (const float*)d_in[16]};
  const float* Wg1[2]  = {(const float*)d_in[12], (const float*)d_in[17]};
  const float* bg1[2]  = {(const float*)d_in[13], (const float*)d_in[18]};
  const float* Wg2[2]  = {(const float*)d_in[14], (const float*)d_in[19]};
  const float* bg2[2]  = {(const float*)d_in[15], (const float*)d_in[20]};
  const float* Wh1 = (const float*)d_in[21]; const float* bh1 = (const float*)d_in[22];
  const float* Wh2 = (const float*)d_in[23]; const float* bh2 = (const float*)d_in[24];

  float* out = (float*)d_out;
  float* ws  = (float*)d_ws;

  // ---- workspace arena (floats); encoder H0/H1 region reused for Gm and HH
  float* H0 = ws;                        // 5,120,000
  float* H1 = ws + 5120000;              // 5,120,000
  float* Gm = ws;                        // 10,240,000 (alias, post-encoder)
  float* HH = ws;                        // alias (head phase)
  float* H  = ws + 10240000;             // 2,560,000
  float* Z  = ws + 12800000;             // 2,560,000
  float* Gh = Z;                         // alias (Z dead after edge_prop)
  float* Mb = ws + 15360000;             // 10,240,000
  float* Sc = ws + 25600000;             // 80,000
  float* Y  = ws + 25680000;             // 2,560,000
  float* St = ws + 28240000;             // 256
  // packed (pair-interleaved) weights
  float* PW = ws + 28240512;
  float* PWi0   = PW;                    // 262144
  float* PWi1   = PW + 262144;           // 65536
  float* PWi2   = PW + 327680;           // 32768
  float* PWh1   = PW + 360448;           // 16384
  float* PWmsg[2] = {PW + 376832, PW + 393216};   // 16384 each
  float* PWg1h[2] = {PW + 409600, PW + 425984};   // top 128 rows of Wg1
  float* PWg1m[2] = {PW + 442368, PW + 458752};   // rows 128..255 of Wg1
  // total ws: 28,715,648 floats = 114.9 MB

  const dim3 blk(256);
  const int gR16 = cdiv_i(N_NODES / 16, 8);             // NCT=16, MT=1
  const int gR32 = cdiv_i(N_NODES / 32, 8);             // NCT=8,  MT=2
  const int gRNK = cdiv_i((N_NODES * K_REL) / 32, 8);   // Gm GEMM

  // ---- pack all weights once (pair-interleave along K)
  pack_w_pairs<<<cdiv_i(512 * 256, 256), blk, 0, stream>>>(Wi0, PWi0, IN_DIM, HID_IN);
  pack_w_pairs<<<cdiv_i(128 * 256, 256), blk, 0, stream>>>(Wi1, PWi1, HID_IN, HID_IN);
  pack_w_pairs<<<cdiv_i(128 * 128, 256), blk, 0, stream>>>(Wi2, PWi2, HID_IN, D_HID);
  pack_w_pairs<<<cdiv_i(64 * 128, 256),  blk, 0, stream>>>(Wh1, PWh1, D_HID, D_HID);
  for (int b = 0; b < 2; ++b) {
    pack_w_pairs<<<cdiv_i(64 * 128, 256), blk, 0, stream>>>(Wmsg[b], PWmsg[b], D_HID, D_HID);
    pack_w_pairs<<<cdiv_i(64 * 128, 256), blk, 0, stream>>>(Wg1[b], PWg1h[b], D_HID, D_HID);
    pack_w_pairs<<<cdiv_i(64 * 128, 256), blk, 0, stream>>>(Wg1[b] + 128 * 128, PWg1m[b], D_HID, D_HID);
  }

  // ---- input encoder: 1024 -> 256 -> 256 -> 128, each + ReLU
  gemm_wmma_f32<16, 1, true, true><<<gR16, blk, 0, stream>>>(X,  PWi0, bi0, H0, N_NODES, IN_DIM);
  gemm_wmma_f32<16, 1, true, true><<<gR16, blk, 0, stream>>>(H0, PWi1, bi1, H1, N_NODES, HID_IN);
  gemm_wmma_f32<8,  2, true, true><<<gR32, blk, 0, stream>>>(H1, PWi2, bi2, H,  N_NODES, HID_IN);

  // ---- two gated message-passing blocks
  for (int b = 0; b < 2; ++b) {
    float* alpha = out + N_NODES + (size_t)b * N_NODES * K_REL;

    // Z = H @ Wmsg
    gemm_wmma_f32<8, 2, false, false><<<gR32, blk, 0, stream>>>(H, PWmsg[b], nullptr, Z, N_NODES, D_HID);

    // M = segment_sum(Z[cols]) over each relation, then clip
    const long nM = (long)N_NODES * K_REL * D_HID;
    zero_f32<<<cdiv_i((int)nM, 256), blk, 0, stream>>>(Mb, nM);
    edge_prop<<<cdiv_i(K_REL * N_EDGE, 8), blk, 0, stream>>>(rows, cols, Z, Mb, N_EDGE);
    clamp_f32<<<cdiv_i((int)nM, 256), blk, 0, stream>>>(Mb, nM, -M_CLAMP, M_CLAMP);

    // gate MLP, decomposed: Gh = H @ Wg1[:128], Gm = M @ Wg1[128:256]
    gemm_wmma_f32<8, 2, false, false><<<gR32, blk, 0, stream>>>(H,  PWg1h[b], nullptr, Gh, N_NODES, D_HID);
    gemm_wmma_f32<8, 2, false, false><<<gRNK, blk, 0, stream>>>(Mb, PWg1m[b], nullptr, Gm, N_NODES * K_REL, D_HID);
    gate_score<<<cdiv_i(N_NODES * K_REL, 8), blk, 0, stream>>>(
        Gh, Gm, logdeg, Wg1[b] + 256 * 128, bg1[b], Wg2[b], bg2[b], Sc);

    // alpha (written straight to output), fuse + residual, PairNorm + ReLU
    softmax_renorm<<<cdiv_i(N_NODES, 256), blk, 0, stream>>>(Sc, mask, alpha);
    fused_residual<<<cdiv_i(N_NODES * D_HID, 256), blk, 0, stream>>>(alpha, Mb, H, Y);
    zero_f32<<<1, blk, 0, stream>>>(St, 136);
    colstats<<<cdiv_i(N_NODES, 256), dim3(128), 0, stream>>>(Y, St, 256);
    finalize_stats<<<1, dim3(128), 0, stream>>>(St);
    norm_relu<<<cdiv_i(N_NODES * D_HID, 256), blk, 0, stream>>>(Y, St, H);
  }

  // ---- head: h = relu(H @ Wh1 + bh1); logits = h @ Wh2 + bh2
  gemm_wmma_f32<8, 2, true, true><<<gR32, blk, 0, stream>>>(H, PWh1, bh1, HH, N_NODES, D_HID);
  dot_out<<<cdiv_i(N_NODES, 8), blk, 0, stream>>>(HH, Wh2, bh2, out);
}